// DNC_43190191129215
// MI455X (gfx1250) — compile-verified
//
#include <hip/hip_runtime.h>
#include <math.h>

// ---- problem constants ----
#define B_    128
#define N_    512
#define W_    128
#define R_    4
#define U_    1024
#define IFACE_ 919
#define EPS_  1e-6f
#define CLIP_ 20.0f

typedef float v2f __attribute__((ext_vector_type(2)));
typedef float v8f __attribute__((ext_vector_type(8)));
typedef int   v4i __attribute__((ext_vector_type(4)));

__device__ __forceinline__ float sig_(float x)  { return 1.0f / (1.0f + expf(-x)); }
__device__ __forceinline__ float sp_(float x)   { return (x > 20.0f) ? x : log1pf(expf(x)); }
__device__ __forceinline__ float clip_(float x) { return fminf(fmaxf(x, -CLIP_), CLIP_); }

__device__ __forceinline__ float waveRed(float v) {
#pragma unroll
  for (int o = 16; o > 0; o >>= 1) v += __shfl_xor(v, o, 32);
  return v;
}

// ---- CDNA5 async global->LDS staging support ----
#if __has_builtin(__builtin_amdgcn_global_load_async_to_lds_b32) && \
    __has_builtin(__builtin_amdgcn_global_load_async_to_lds_b128)
#define HAVE_ASYNC_LDS 1
#else
#define HAVE_ASYNC_LDS 0
#endif

// typed address-space casts matching the builtin signatures
// (param1: AS1 "global/__device__" pointer, param2: AS3 LDS pointer)
#define GPTR32(p)  ((__attribute__((address_space(1))) int*)(p))
#define GPTR128(p) ((__attribute__((address_space(1))) v4i*)(p))
#define LPTR32(p)  ((__attribute__((address_space(3))) int*)(p))
#define LPTR128(p) ((__attribute__((address_space(3))) v4i*)(p))

#if __has_builtin(__builtin_amdgcn_s_wait_asynccnt)
#define WAIT_ASYNC(n) __builtin_amdgcn_s_wait_asynccnt(n)
#else
#define WAIT_ASYNC(n) asm volatile("s_wait_asynccnt %0" :: "i"(n) : "memory")
#endif

// =====================================================================
// fp32 WMMA GEMM: C(M,N) = A(M,K) @ B(K,N) (+bias, optional clip)
// B rows < ksplit come from B0, rows >= ksplit from B1 (concat weights).
// 64x64 tile/block; 8 waves x two 16x16 subtiles (shared B fragment).
// KC=32 K-chunks double-buffered in LDS via GLOBAL_LOAD_ASYNC_TO_LDS_*
// (ASYNCcnt; in-order completion => s_wait_asynccnt P releases old tile).
// Fragment layout per ISA 7.12.2 (32-bit A 16x4 / B 4x16):
//   lane<16 : (k0,k0+1) of row (lane);  lane>=16 : (k0+2,k0+3) of row-16
// =====================================================================
#define KC_  32
#define LDT_ 36   // stride: mult of 4 -> 16B-aligned b128 stores; 16 distinct banks

__global__ __launch_bounds__(256) void gemm_f32_wmma(
    const float* __restrict__ A, int lda,
    const float* __restrict__ B0, const float* __restrict__ B1, int ksplit, int ldb,
    const float* __restrict__ bias,
    float* __restrict__ C, int ldc,
    int M, int N, int K, int do_clip)
{
#if HAVE_ASYNC_LDS
  __shared__ float As[2][64 * LDT_];
  __shared__ float Bs[2][64 * LDT_];
#else
  __shared__ float As[1][64 * LDT_];
  __shared__ float Bs[1][64 * LDT_];
#endif

  const int t    = threadIdx.x;
  const int lane = t & 31;
  const int wave = t >> 5;
  const int N0 = blockIdx.x * 64;
  const int M0 = blockIdx.y * 64;

  const int mi0 = wave >> 2;       // 0..1 ; second subtile at mi0+2
  const int ni  = wave & 3;        // 0..3
  const int lh  = lane >> 4;       // 0/1 -> k offset 0/2
  const int lm  = lane & 15;

  v8f acc0 = {0.f,0.f,0.f,0.f,0.f,0.f,0.f,0.f};
  v8f acc1 = {0.f,0.f,0.f,0.f,0.f,0.f,0.f,0.f};

  const int arow0 = (mi0 * 16 + lm) * LDT_;
  const int arow1 = ((mi0 + 2) * 16 + lm) * LDT_;
  const int brow  = (ni * 16 + lm) * LDT_;

  // A-tile: 64x32 floats as 512 float4s -> 2 per thread (q: m = q/8, kq = q%8)
  // B-tile: 32x64 floats transposed to Bs[n][k], 8 scalars per thread.
  // OOB N columns: clamp global column (dup data feeds only never-stored cols).

  auto compute = [&](const float* as, const float* bs) {
#pragma unroll
    for (int kk = 0; kk < KC_; kk += 4) {
      const int ko = kk + lh * 2;
      v2f bfrag = *(const v2f*)&bs[brow + ko];
      v2f a0    = *(const v2f*)&as[arow0 + ko];
      v2f a1    = *(const v2f*)&as[arow1 + ko];
      acc0 = __builtin_amdgcn_wmma_f32_16x16x4_f32(false, a0, false, bfrag,
                                                   (short)0, acc0, false, false);
      acc1 = __builtin_amdgcn_wmma_f32_16x16x4_f32(false, a1, false, bfrag,
                                                   (short)0, acc1, false, false);
    }
  };

#if HAVE_ASYNC_LDS
  auto stage = [&](int buf, int kt) {
#pragma unroll
    for (int j = 0; j < 2; ++j) {                 // A: 2 x b128 per thread
      int q = t + j * 256;
      int m = q >> 3, kq = q & 7;
      const float* g = &A[(size_t)(M0 + m) * lda + kt + kq * 4];
      __builtin_amdgcn_global_load_async_to_lds_b128(
          GPTR128(g), LPTR128(&As[buf][m * LDT_ + kq * 4]), 0, 0);
    }
#pragma unroll
    for (int j = 0; j < 8; ++j) {                 // B: 8 x b32 per thread (transpose)
      int e = t + j * 256;
      int k = e >> 6, n = e & 63;
      int col = N0 + n; if (col >= N) col = N - 1;
      int row = kt + k;
      const float* g = (row < ksplit) ? &B0[(size_t)row * ldb + col]
                                      : &B1[(size_t)(row - ksplit) * ldb + col];
      __builtin_amdgcn_global_load_async_to_lds_b32(
          GPTR32(g), LPTR32(&Bs[buf][n * LDT_ + k]), 0, 0);
    }
  };

  const int nT = K / KC_;
  stage(0, 0);
  for (int it = 0; it < nT; ++it) {
    const int buf = it & 1;
    if (it + 1 < nT) {
      stage(buf ^ 1, (it + 1) * KC_);
      WAIT_ASYNC(10);             // 10 newer ops outstanding -> tile `it` complete
    } else {
      WAIT_ASYNC(0);
    }
    __syncthreads();              // all waves' staging of `buf` visible
    compute(As[buf], Bs[buf]);
    __syncthreads();              // done reading `buf` before it is restaged
  }
#else
  for (int kt = 0; kt < K; kt += KC_) {
    float4 av[2]; float bv[8];
#pragma unroll
    for (int j = 0; j < 2; ++j) {
      int q = t + j * 256;
      int m = q >> 3, kq = q & 7;
      av[j] = *(const float4*)&A[(size_t)(M0 + m) * lda + kt + kq * 4];
    }
#pragma unroll
    for (int j = 0; j < 8; ++j) {
      int e = t + j * 256;
      int k = e >> 6, n = e & 63;
      int col = N0 + n; if (col >= N) col = N - 1;
      int row = kt + k;
      bv[j] = (row < ksplit) ? B0[(size_t)row * ldb + col]
                             : B1[(size_t)(row - ksplit) * ldb + col];
    }
#pragma unroll
    for (int j = 0; j < 2; ++j) {
      int q = t + j * 256;
      int m = q >> 3, kq = q & 7;
      *(float4*)&As[0][m * LDT_ + kq * 4] = av[j];
    }
#pragma unroll
    for (int j = 0; j < 8; ++j) {
      int e = t + j * 256;
      int k = e >> 6, n = e & 63;
      Bs[0][n * LDT_ + k] = bv[j];
    }
    __syncthreads();
    compute(As[0], Bs[0]);
    __syncthreads();
  }
#endif

  // epilogue: C/D layout — VGPR v, lane l: row = v + 8*(l/16), col = l%16
  const int col = N0 + ni * 16 + lm;
  if (col < N) {
    const float bv = bias ? bias[col] : 0.f;
#pragma unroll
    for (int v = 0; v < 8; v++) {
      int r0 = M0 + mi0 * 16 + v + 8 * lh;
      int r1 = M0 + (mi0 + 2) * 16 + v + 8 * lh;
      float x0 = acc0[v] + bv;
      float x1 = acc1[v] + bv;
      if (do_clip) { x0 = clip_(x0); x1 = clip_(x1); }
      C[r0 * ldc + col] = x0;
      C[r1 * ldc + col] = x1;
    }
  }
}

// ---- pack xh = [inputs | read_vectors_flat | h]  (128 x 2048) ----
__global__ void pack_xh(const float* __restrict__ inp, const float* __restrict__ rv,
                        const float* __restrict__ h, float* __restrict__ xh)
{
  int i = blockIdx.x * 256 + threadIdx.x;
  if (i >= B_ * 2048) return;
  int b = i >> 11, col = i & 2047;
  float v;
  if (col < 512)       v = inp[b * 512 + col];
  else if (col < 1024) v = rv[b * 512 + (col - 512)];
  else                 v = h[b * U_ + (col - 1024)];
  xh[i] = v;
}

// ---- LSTM elementwise ----
__global__ void lstm_elem(const float* __restrict__ z, const float* __restrict__ c,
                          float* __restrict__ h_new, float* __restrict__ c_new)
{
  int i = blockIdx.x * 256 + threadIdx.x;
  if (i >= B_ * U_) return;
  int b = i >> 10, u = i & 1023;
  const float* zr = z + (size_t)b * 4096;
  float ig = sig_(zr[u]);
  float fg = sig_(zr[1024 + u]);
  float zc = zr[2048 + u];
  float og = sig_(zr[3072 + u]);
  float cn = fg * c[i] + ig * tanhf(zc);
  float hn = og * tanhf(cn);
  c_new[i] = cn;
  h_new[i] = hn;   // ctrl = clip(h_new) == h_new (|h_new|<1)
}

// ---- interface transforms (per batch), iface stride 416 floats:
// [0..3]=rstr, [4]=wstr, [5]=ag, [6]=wg, [8..135]=wkey, [136..263]=erase,
// [264..391]=wvec, [392..395]=fg, [396..407]=modes[m*4+r]
__global__ void iface_elem(const float* __restrict__ iv, float* __restrict__ ifc)
{
  int b = blockIdx.x, t = threadIdx.x;
  const float* r = iv + (size_t)b * IFACE_;
  float* f = ifc + (size_t)b * 416;
  if (t < 128) {
    f[8 + t]   = r[516 + t];          // write_key
    f[136 + t] = sig_(r[645 + t]);    // erase
    f[264 + t] = r[773 + t];          // write_vec
  }
  if (t < 4) {
    f[t]       = 1.0f + sp_(r[512 + t]);   // read_strengths
    f[392 + t] = sig_(r[901 + t]);         // free_gates
    float m0 = r[907 + t], m1 = r[911 + t], m2 = r[915 + t];
    float mx = fmaxf(m0, fmaxf(m1, m2));
    float e0 = expf(m0 - mx), e1 = expf(m1 - mx), e2 = expf(m2 - mx);
    float s = e0 + e1 + e2;
    f[396 + t] = e0 / s; f[400 + t] = e1 / s; f[404 + t] = e2 / s;
  }
  if (t == 0) {
    f[4] = 1.0f + sp_(r[644]);  // write_strength
    f[5] = sig_(r[905]);        // alloc_gate
    f[6] = sig_(r[906]);        // write_gate
  }
}

// ---- psi & usage_new ----
__global__ void usage_elem(const float* __restrict__ ifc, const float* __restrict__ read_w,
                           const float* __restrict__ usage, const float* __restrict__ write_w,
                           float* __restrict__ usage_new)
{
  int i = blockIdx.x * 256 + threadIdx.x;
  if (i >= B_ * N_) return;
  int b = i >> 9;
  const float* fb = ifc + (size_t)b * 416;
  const float4 rw = *(const float4*)&read_w[(size_t)i * 4];
  float psi = (1.f - fb[392] * rw.x) * (1.f - fb[393] * rw.y) *
              (1.f - fb[394] * rw.z) * (1.f - fb[395] * rw.w);
  float u = usage[i], w = write_w[i];
  usage_new[i] = (u + w - u * w) * psi;
}

// ---- bitonic sort usage_new -> alloc weighting (one block per batch) ----
__global__ __launch_bounds__(256) void sort_alloc(const float* __restrict__ usage_new,
                                                  float* __restrict__ alloc)
{
  __shared__ float sk[512];
  __shared__ int   si[512];
  __shared__ float cp[512];
  int b = blockIdx.x, t = threadIdx.x;
  sk[t]       = usage_new[b * N_ + t];        si[t]       = t;
  sk[t + 256] = usage_new[b * N_ + t + 256];  si[t + 256] = t + 256;
  __syncthreads();
  for (int k = 2; k <= 512; k <<= 1) {
    for (int j = k >> 1; j > 0; j >>= 1) {
      int i   = ((t & ~(j - 1)) << 1) | (t & (j - 1));
      int ixj = i | j;
      bool up = ((i & k) == 0);
      float a = sk[i], bv = sk[ixj];
      int ia = si[i], ib = si[ixj];
      bool gt = (a > bv) || (a == bv && ia > ib);   // stable tiebreak
      if (gt == up) { sk[i] = bv; sk[ixj] = a; si[i] = ib; si[ixj] = ia; }
      __syncthreads();
    }
  }
  if (t == 0) {                       // exclusive cumprod of sorted usage
    float p = 1.f;
    for (int s = 0; s < 512; s++) { cp[s] = p; p *= sk[s]; }
  }
  __syncthreads();
  alloc[b * N_ + si[t]]       = (1.f - sk[t])       * cp[t];
  alloc[b * N_ + si[t + 256]] = (1.f - sk[t + 256]) * cp[t + 256];
}

// ---- write content weights, ww, precedence_new (one block per batch) ----
__global__ __launch_bounds__(256) void wc_ww_prec(
    const float* __restrict__ memory, const float* __restrict__ ifc,
    const float* __restrict__ alloc, const float* __restrict__ precedence,
    float* __restrict__ ww_out, float* __restrict__ prec_out)
{
  __shared__ float wkey[128];
  __shared__ float sc[512];
  __shared__ float wwl[512];
  __shared__ float red[256];
  __shared__ float knS, mxS, smS, sumWW;
  int b = blockIdx.x, t = threadIdx.x;
  int lane = t & 31, wave = t >> 5;
  const float* f = ifc + (size_t)b * 416;
  if (t < 128) wkey[t] = f[8 + t];
  __syncthreads();
  red[t] = (t < 128) ? wkey[t] * wkey[t] : 0.f;
  __syncthreads();
  for (int s = 128; s > 0; s >>= 1) { if (t < s) red[t] += red[t + s]; __syncthreads(); }
  if (t == 0) knS = sqrtf(red[0]);
  __syncthreads();
  const float kn = knS, wstr = f[4], ag = f[5], wg = f[6];
  for (int row = wave; row < N_; row += 8) {
    float dot = 0.f, nrm = 0.f;
    const float* mrow = memory + ((size_t)b * N_ + row) * W_;
    for (int w = lane; w < W_; w += 32) {
      float m = mrow[w];
      dot += m * wkey[w];
      nrm += m * m;
    }
    dot = waveRed(dot); nrm = waveRed(nrm);
    if (lane == 0) sc[row] = wstr * (dot / (sqrtf(nrm) * kn + EPS_));
  }
  __syncthreads();
  red[t] = fmaxf(sc[t], sc[t + 256]);
  __syncthreads();
  for (int s = 128; s > 0; s >>= 1) { if (t < s) red[t] = fmaxf(red[t], red[t + s]); __syncthreads(); }
  if (t == 0) mxS = red[0];
  __syncthreads();
  float e0 = expf(sc[t] - mxS), e1 = expf(sc[t + 256] - mxS);
  red[t] = e0 + e1;
  __syncthreads();
  for (int s = 128; s > 0; s >>= 1) { if (t < s) red[t] += red[t + s]; __syncthreads(); }
  if (t == 0) smS = red[0];
  __syncthreads();
  float wc0 = e0 / smS, wc1 = e1 / smS;
  float w0 = wg * (ag * alloc[b * N_ + t]       + (1.f - ag) * wc0);
  float w1 = wg * (ag * alloc[b * N_ + t + 256] + (1.f - ag) * wc1);
  wwl[t] = w0; wwl[t + 256] = w1;
  ww_out[b * N_ + t] = w0; ww_out[b * N_ + t + 256] = w1;
  red[t] = w0 + w1;
  __syncthreads();
  for (int s = 128; s > 0; s >>= 1) { if (t < s) red[t] += red[t + s]; __syncthreads(); }
  if (t == 0) sumWW = red[0];
  __syncthreads();
  prec_out[b * N_ + t]       = (1.f - sumWW) * precedence[b * N_ + t]       + wwl[t];
  prec_out[b * N_ + t + 256] = (1.f - sumWW) * precedence[b * N_ + t + 256] + wwl[t + 256];
}

// ---- memory update ----
__global__ void mem_update(const float* __restrict__ memory, const float* __restrict__ ww,
                           const float* __restrict__ ifc, float* __restrict__ memory_new)
{
  int i = blockIdx.x * 256 + threadIdx.x;
  if (i >= B_ * N_ * W_) return;
  int b = i >> 16;
  int n = (i >> 7) & 511;
  int w = i & 127;
  const float* f = ifc + (size_t)b * 416;
  float wwv = ww[b * N_ + n];
  memory_new[i] = memory[i] * (1.f - wwv * f[136 + w]) + wwv * f[264 + w];
}

// ---- link_new + fw (fused; one link read, one write) ----
__global__ __launch_bounds__(256) void link_fw(
    const float* __restrict__ link, const float* __restrict__ ww,
    const float* __restrict__ precedence, const float* __restrict__ read_w,
    float* __restrict__ link_new, float* __restrict__ fw)
{
  __shared__ float rwl[2048];
  __shared__ float wwl[512];
  __shared__ float pl[512];
  int b = blockIdx.y, i0 = blockIdx.x * 64;
  int t = threadIdx.x, lane = t & 31, wave = t >> 5;
  for (int i = t; i < 2048; i += 256) rwl[i] = read_w[(size_t)b * 2048 + i];
  for (int i = t; i < 512; i += 256) {
    wwl[i] = ww[b * N_ + i];
    pl[i]  = precedence[b * N_ + i];
  }
  __syncthreads();
  for (int ii = wave; ii < 64; ii += 8) {
    int i = i0 + ii;
    float wwi = wwl[i];
    const float* lrow = link + ((size_t)b * N_ + i) * N_;
    float* orow = link_new + ((size_t)b * N_ + i) * N_;
    float f0 = 0.f, f1 = 0.f, f2 = 0.f, f3 = 0.f;
    for (int j = lane; j < N_; j += 32) {
      float ln = (1.f - wwi - wwl[j]) * lrow[j] + wwi * pl[j];
      if (j == i) ln = 0.f;
      orow[j] = ln;
      f0 += ln * rwl[j * 4 + 0];
      f1 += ln * rwl[j * 4 + 1];
      f2 += ln * rwl[j * 4 + 2];
      f3 += ln * rwl[j * 4 + 3];
    }
    f0 = waveRed(f0); f1 = waveRed(f1); f2 = waveRed(f2); f3 = waveRed(f3);
    if (lane == 0) {
      float* fo = fw + ((size_t)b * N_ + i) * 4;
      fo[0] = f0; fo[1] = f1; fo[2] = f2; fo[3] = f3;
    }
  }
}

// ---- bw = link_new^T @ read_w (column-oriented, coalesced) ----
__global__ __launch_bounds__(256) void bw_kernel(
    const float* __restrict__ link_new, const float* __restrict__ read_w,
    float* __restrict__ bw)
{
  __shared__ float rwl[2048];
  int b = blockIdx.y;
  int i = blockIdx.x * 256 + threadIdx.x;
  for (int k = threadIdx.x; k < 2048; k += 256) rwl[k] = read_w[(size_t)b * 2048 + k];
  __syncthreads();
  float a0 = 0.f, a1 = 0.f, a2 = 0.f, a3 = 0.f;
  const float* base = link_new + (size_t)b * N_ * N_ + i;
  for (int j = 0; j < N_; j++) {
    float v = base[(size_t)j * N_];
    a0 += v * rwl[j * 4 + 0];
    a1 += v * rwl[j * 4 + 1];
    a2 += v * rwl[j * 4 + 2];
    a3 += v * rwl[j * 4 + 3];
  }
  float* o = bw + ((size_t)b * N_ + i) * 4;
  o[0] = a0; o[1] = a1; o[2] = a2; o[3] = a3;
}

// ---- read content scores (wave per row) ----
__global__ __launch_bounds__(256) void rc_scores(
    const float* __restrict__ memory_new, const float* __restrict__ iv,
    const float* __restrict__ ifc, float* __restrict__ scores)
{
  __shared__ float rk[512];
  __shared__ float kn[4];
  __shared__ float rstr[4];
  int b = blockIdx.y, i0 = blockIdx.x * 64;
  int t = threadIdx.x, lane = t & 31, wave = t >> 5;
  rk[t]       = iv[(size_t)b * IFACE_ + t];
  rk[t + 256] = iv[(size_t)b * IFACE_ + t + 256];
  if (t < 4) rstr[t] = ifc[(size_t)b * 416 + t];
  __syncthreads();
  if (t < 4) {
    float s = 0.f;
    for (int w = 0; w < 128; w++) { float v = rk[w * 4 + t]; s += v * v; }
    kn[t] = sqrtf(s);
  }
  __syncthreads();
  for (int ii = wave; ii < 64; ii += 8) {
    int i = i0 + ii;
    const float* mrow = memory_new + ((size_t)b * N_ + i) * W_;
    float d0 = 0.f, d1 = 0.f, d2 = 0.f, d3 = 0.f, nrm = 0.f;
    for (int w = lane; w < W_; w += 32) {
      float m = mrow[w];
      nrm += m * m;
      d0 += m * rk[w * 4 + 0];
      d1 += m * rk[w * 4 + 1];
      d2 += m * rk[w * 4 + 2];
      d3 += m * rk[w * 4 + 3];
    }
    d0 = waveRed(d0); d1 = waveRed(d1); d2 = waveRed(d2); d3 = waveRed(d3);
    nrm = waveRed(nrm);
    if (lane == 0) {
      float mn = sqrtf(nrm);
      float* o = scores + ((size_t)b * N_ + i) * 4;
      o[0] = rstr[0] * (d0 / (mn * kn[0] + EPS_));
      o[1] = rstr[1] * (d1 / (mn * kn[1] + EPS_));
      o[2] = rstr[2] * (d2 / (mn * kn[2] + EPS_));
      o[3] = rstr[3] * (d3 / (mn * kn[3] + EPS_));
    }
  }
}

// ---- softmax over N per (b,r) + combine read modes -> rw ----
__global__ __launch_bounds__(256) void softmax_rw(
    const float* __restrict__ scores, const float* __restrict__ ifc,
    const float* __restrict__ fw, const float* __restrict__ bw,
    float* __restrict__ rw_out)
{
  __shared__ float red[256];
  __shared__ float mxS, smS;
  int b = blockIdx.x >> 2, r = blockIdx.x & 3;
  int t = threadIdx.x;
  size_t base = (size_t)b * N_ * 4 + r;
  float s0 = scores[base + (size_t)t * 4];
  float s1 = scores[base + (size_t)(t + 256) * 4];
  red[t] = fmaxf(s0, s1);
  __syncthreads();
  for (int s = 128; s > 0; s >>= 1) { if (t < s) red[t] = fmaxf(red[t], red[t + s]); __syncthreads(); }
  if (t == 0) mxS = red[0];
  __syncthreads();
  float e0 = expf(s0 - mxS), e1 = expf(s1 - mxS);
  red[t] = e0 + e1;
  __syncthreads();
  for (int s = 128; s > 0; s >>= 1) { if (t < s) red[t] += red[t + s]; __syncthreads(); }
  if (t == 0) smS = red[0];
  __syncthreads();
  const float* f = ifc + (size_t)b * 416;
  float m0 = f[396 + r], m1 = f[400 + r], m2 = f[404 + r];
  rw_out[base + (size_t)t * 4] =
      m0 * bw[base + (size_t)t * 4] + m1 * (e0 / smS) + m2 * fw[base + (size_t)t * 4];
  rw_out[base + (size_t)(t + 256) * 4] =
      m0 * bw[base + (size_t)(t + 256) * 4] + m1 * (e1 / smS) + m2 * fw[base + (size_t)(t + 256) * 4];
}

// ---- read_vectors_new = memory_new^T @ rw (per batch; thread = w) ----
__global__ __launch_bounds__(128) void rvn_kernel(
    const float* __restrict__ memory_new, const float* __restrict__ rw,
    float* __restrict__ rvn)
{
  __shared__ float rwl[2048];
  int b = blockIdx.x, w = threadIdx.x;
  for (int i = w; i < 2048; i += 128) rwl[i] = rw[(size_t)b * 2048 + i];
  __syncthreads();
  float a0 = 0.f, a1 = 0.f, a2 = 0.f, a3 = 0.f;
  const float* base = memory_new + (size_t)b * N_ * W_ + w;
  for (int n = 0; n < N_; n++) {
    float m = base[(size_t)n * W_];
    a0 += m * rwl[n * 4 + 0];
    a1 += m * rwl[n * 4 + 1];
    a2 += m * rwl[n * 4 + 2];
    a3 += m * rwl[n * 4 + 3];
  }
  float* o = rvn + (size_t)b * 512 + w * 4;
  o[0] = a0; o[1] = a1; o[2] = a2; o[3] = a3;
}

// ---- pack xo = [clip(h_new) | read_vectors_new_flat]  (128 x 1536) ----
__global__ void pack_xo(const float* __restrict__ h_new, const float* __restrict__ rvn,
                        float* __restrict__ xo)
{
  int i = blockIdx.x * 256 + threadIdx.x;
  if (i >= B_ * 1536) return;
  int b = i / 1536, col = i % 1536;
  xo[i] = (col < 1024) ? clip_(h_new[b * U_ + col]) : rvn[b * 512 + (col - 1024)];
}

// =====================================================================
extern "C" void kernel_launch(void* const* d_in, const int* in_sizes, int n_in,
                              void* d_out, int out_size, void* d_ws, size_t ws_size,
                              hipStream_t stream) {
  const float* inputs       = (const float*)d_in[0];
  const float* memory       = (const float*)d_in[1];
  const float* usage        = (const float*)d_in[2];
  const float* precedence   = (const float*)d_in[3];
  const float* link         = (const float*)d_in[4];
  const float* write_w      = (const float*)d_in[5];
  const float* read_w       = (const float*)d_in[6];
  const float* h            = (const float*)d_in[7];
  const float* c            = (const float*)d_in[8];
  const float* read_vectors = (const float*)d_in[9];
  const float* lstm_kernel  = (const float*)d_in[10];
  const float* lstm_rkernel = (const float*)d_in[11];
  const float* lstm_bias    = (const float*)d_in[12];
  const float* Wi           = (const float*)d_in[13];
  const float* bi           = (const float*)d_in[14];
  const float* Wo           = (const float*)d_in[15];
  const float* bo           = (const float*)d_in[16];

  float* out = (float*)d_out;
  float* out_o   = out;                       // (B,OUT)        65536
  float* mem_o   = out_o   + 65536;           // (B,N,W)        8388608
  float* usage_o = mem_o   + 8388608;         // (B,N)          65536
  float* prec_o  = usage_o + 65536;           // (B,N)          65536
  float* link_o  = prec_o  + 65536;           // (B,N,N)        33554432
  float* ww_o    = link_o  + 33554432;        // (B,N)          65536
  float* rw_o    = ww_o    + 65536;           // (B,N,R)        262144
  float* h_o     = rw_o    + 262144;          // (B,U)          131072
  float* c_o     = h_o     + 131072;          // (B,U)          131072
  float* rvn_o   = c_o     + 131072;          // (B,W,R)        65536

  float* ws = (float*)d_ws;                   // needs ~8 MB
  float* XH    = ws;                          // 128*2048
  float* Z     = XH    + 262144;              // 128*4096
  float* IV    = Z     + 524288;              // 128*919
  float* IFC   = IV    + 117632;              // 128*416
  float* ALLOC = IFC   + 53248;               // 128*512
  float* FW    = ALLOC + 65536;               // 128*512*4
  float* BW    = FW    + 262144;              // 128*512*4
  float* SCORE = BW    + 262144;              // 128*512*4
  float* XO    = SCORE + 262144;              // 128*1536

  // 1) pack [x | rv | h]
  pack_xh<<<(B_ * 2048 + 255) / 256, 256, 0, stream>>>(inputs, read_vectors, h, XH);
  // 2) z = XH @ [lstm_kernel; lstm_rkernel] + bias   (M=128,N=4096,K=2048)
  gemm_f32_wmma<<<dim3(64, 2), 256, 0, stream>>>(XH, 2048, lstm_kernel, lstm_rkernel,
                                                 1024, 4096, lstm_bias, Z, 4096,
                                                 128, 4096, 2048, 0);
  // 3) LSTM gates
  lstm_elem<<<(B_ * U_ + 255) / 256, 256, 0, stream>>>(Z, c, h_o, c_o);
  // 4) iv = ctrl @ Wi + bi   (M=128,N=919,K=1024)
  gemm_f32_wmma<<<dim3(15, 2), 256, 0, stream>>>(h_o, 1024, Wi, Wi, 1024, IFACE_, bi,
                                                 IV, IFACE_, 128, IFACE_, 1024, 0);
  // 5) interface activations
  iface_elem<<<B_, 128, 0, stream>>>(IV, IFC);
  // 6) psi & usage_new
  usage_elem<<<(B_ * N_ + 255) / 256, 256, 0, stream>>>(IFC, read_w, usage, write_w, usage_o);
  // 7) sort + allocation weighting
  sort_alloc<<<B_, 256, 0, stream>>>(usage_o, ALLOC);
  // 8) write content weights, ww, precedence_new
  wc_ww_prec<<<B_, 256, 0, stream>>>(memory, IFC, ALLOC, precedence, ww_o, prec_o);
  // 9) memory_new
  mem_update<<<(B_ * N_ * W_ + 255) / 256, 256, 0, stream>>>(memory, ww_o, IFC, mem_o);
  // 10) link_new + fw (fused)
  link_fw<<<dim3(8, B_), 256, 0, stream>>>(link, ww_o, precedence, read_w, link_o, FW);
  // 11) bw
  bw_kernel<<<dim3(2, B_), 256, 0, stream>>>(link_o, read_w, BW);
  // 12) read content scores
  rc_scores<<<dim3(8, B_), 256, 0, stream>>>(mem_o, IV, IFC, SCORE);
  // 13) softmax + mode mix -> rw
  softmax_rw<<<B_ * R_, 256, 0, stream>>>(SCORE, IFC, FW, BW, rw_o);
  // 14) read_vectors_new
  rvn_kernel<<<B_, 128, 0, stream>>>(mem_o, rw_o, rvn_o);
  // 15) pack [ctrl | rvn]
  pack_xo<<<(B_ * 1536 + 255) / 256, 256, 0, stream>>>(h_o, rvn_o, XO);
  // 16) out = XO @ Wo + bo, clipped   (M=128,N=512,K=1536)
  gemm_f32_wmma<<<dim3(8, 2), 256, 0, stream>>>(XO, 1536, Wo, Wo, 1536, 512, bo,
                                                out_o, 512, 128, 512, 1536, 1);
}